// VectorQuantizer_33148557590734
// MI455X (gfx1250) — compile-verified
//
#include <hip/hip_runtime.h>
#include <math.h>

typedef float v2f __attribute__((ext_vector_type(2)));
typedef float v8f __attribute__((ext_vector_type(8)));
typedef unsigned int u32x4 __attribute__((ext_vector_type(4)));
typedef int i32x4 __attribute__((ext_vector_type(4)));
typedef int i32x8 __attribute__((ext_vector_type(8)));

#define N_ROWS   8192
#define N_E      8192
#define E_DIM    512
#define MU       0.25f
#define BETA     1.0f
#define BIG_NEG  1.0e12f

#define WG_ROWS  64          // latent rows per workgroup = 4 row-tiles of 16
#define PANEL    32          // codebook rows per LDS panel = 2 col-tiles of 16
#define LSTR     516         // padded LDS stride (floats); 516 % 64 == 4 -> conflict-free
// LDS: A[64][516] + two B panels [32][516] each (TDM double buffer) + 512 reduce floats
#define SMEM_FLOATS ((WG_ROWS + 2 * PANEL) * LSTR + 512)
#define SMEM_BYTES  (SMEM_FLOATS * sizeof(float))

// ---------------------------------------------------------------------------
// Tensor Data Mover: async DMA of a [rows x 512] f32 tile into LDS.
// TDM padding (2 DWORDs after every 256 DWORDs) reproduces the 516-float
// bank-conflict-free stride in hardware: element (r,k) lands at
// lds + r*516 + k + 2*(k>=256)  (floats).
// ---------------------------------------------------------------------------
__device__ __forceinline__ void tdm_load_panel(void* ldsDst, const void* gsrc, int rows)
{
  unsigned lds_addr = (unsigned)(size_t)ldsDst;              // LDS aperture: low 32 bits = offset
  unsigned long long ga = (unsigned long long)(size_t)gsrc;

  u32x4 g0;
  g0[0] = 1u;                                  // count=1 valid descriptor
  g0[1] = lds_addr;                            // lds_addr (bytes)
  g0[2] = (unsigned)ga;                        // global_addr[31:0]
  g0[3] = (unsigned)(ga >> 32) | 0x80000000u;  // global_addr[56:32] | type=2<<30

  i32x8 g1;
  g1[0] = (2 << 16)      // data_size = 4 bytes
        | (1 << 20)      // pad_enable
        | (7 << 22)      // pad_interval: every 256 DWORDs
        | (1 << 25);     // pad_amount: 2 DWORDs
  g1[1] = (int)(512u << 16);                   // tensor_dim0 = 512   (bits 79:48)
  g1[2] = (int)(8192u << 16);                  // tensor_dim1 = 8192  (bits 111:80)
  g1[3] = (int)(512u << 16);                   // tile_dim0 = 512     (bits 127:112)
  g1[4] = rows;                                // tile_dim1 = rows    (bits 143:128)
  g1[5] = 512;                                 // tensor_dim0_stride = 512 (bits 207:160)
  g1[6] = 0;
  g1[7] = 0;

  i32x4 gz = {0, 0, 0, 0};
#if __clang_major__ >= 23
  i32x8 gz8 = {0, 0, 0, 0, 0, 0, 0, 0};
  __builtin_amdgcn_tensor_load_to_lds(g0, g1, gz, gz, gz8, 0);
#else
  __builtin_amdgcn_tensor_load_to_lds(g0, g1, gz, gz, 0);
#endif
}

// ---------------------------------------------------------------------------
// GEMM1: distances + argmin.  d(i,j) = en[j] - 2*dot(z_i, e_j)  (||z||^2 is
// constant per row -> dropped).  f32 WMMA 16x16x4, K = 512, TDM-fed double-
// buffered B panels.
// ---------------------------------------------------------------------------
__global__ __launch_bounds__(256, 1)
void vq_dist_argmin(const float* __restrict__ latent,
                    const float* __restrict__ emb,
                    const float* __restrict__ en,
                    int* __restrict__ outIdx)
{
  extern __shared__ float smem[];
  float* sA  = smem;                                 // [WG_ROWS][LSTR]
  float* sB0 = smem + WG_ROWS * LSTR;                // [PANEL][LSTR]  TDM buffer 0
  float* sB1 = sB0 + PANEL * LSTR;                   // [PANEL][LSTR]  TDM buffer 1
  float* sR  = sB1 + PANEL * LSTR;                   // reduce scratch (512 floats)

  const int tid  = threadIdx.x;
  const int lane = tid & 31;
  const int wave = tid >> 5;
  const int lo   = lane & 15;
  const int hi   = lane >> 4;
  const int rt   = wave >> 1;   // row tile 0..3
  const int ch   = wave &  1;   // column half 0..1
  const int row0 = blockIdx.x * WG_ROWS;

  // kick off TDM for panel 0 while we stage A manually
  if (wave == 0) tdm_load_panel(sB0, emb, PANEL);

  for (int t = tid; t < WG_ROWS * (E_DIM / 4); t += 256) {
    int r  = t >> 7;
    int c4 = t & 127;
    float4 v = *(const float4*)(latent + (size_t)(row0 + r) * E_DIM + c4 * 4);
    *(float4*)(sA + r * LSTR + c4 * 4) = v;
  }

  float best[8];
  int   bidx[8];
#pragma unroll
  for (int v = 0; v < 8; ++v) { best[v] = 3.4e38f; bidx[v] = 0; }

  const float* aBase  = sA  + (rt * 16 + lo) * LSTR + 2 * hi;
  const float* bBase0 = sB0 + (ch * 16 + lo) * LSTR + 2 * hi;
  const float* bBase1 = sB1 + (ch * 16 + lo) * LSTR + 2 * hi;

  if (wave == 0) __builtin_amdgcn_s_wait_tensorcnt(0);
  __syncthreads();                                   // A staged, panel 0 in LDS

  int cur = 0;
  for (int cb = 0; cb < N_E; cb += PANEL) {
    if (wave == 0 && cb + PANEL < N_E)               // DMA next panel into other buffer
      tdm_load_panel(cur ? (void*)sB0 : (void*)sB1,
                     emb + (size_t)(cb + PANEL) * E_DIM, PANEL);

    const float* bB = cur ? bBase1 : bBase0;
    v8f acc0 = {}; v8f acc1 = {};                    // two chains for WMMA ILP
#pragma unroll 8
    for (int kk = 0; kk < E_DIM; kk += 8) {
      const int bo = kk + ((kk >= 256) ? 2 : 0);     // TDM pad gap at mid-row
      v2f a0 = *(const v2f*)(aBase + kk);
      v2f b0 = *(const v2f*)(bB + bo);
      v2f a1 = *(const v2f*)(aBase + kk + 4);
      v2f b1 = *(const v2f*)(bB + bo + 4);
      acc0 = __builtin_amdgcn_wmma_f32_16x16x4_f32(false, a0, false, b0,
                                                   (short)0, acc0, false, false);
      acc1 = __builtin_amdgcn_wmma_f32_16x16x4_f32(false, a1, false, b1,
                                                   (short)0, acc1, false, false);
    }

    const int   col = cb + ch * 16 + lo;             // C/D layout: n = lane%16
    const float ec  = en[col];
#pragma unroll
    for (int v = 0; v < 8; ++v) {                    // row m = v + 8*(lane/16)
      float d = ec - 2.0f * (acc0[v] + acc1[v]);
      if (d < best[v]) { best[v] = d; bidx[v] = col; }
    }

    if (wave == 0) __builtin_amdgcn_s_wait_tensorcnt(0);
    __syncthreads();                                 // next panel ready, reads done
    cur ^= 1;
  }

  // argmin across the 16 lanes of each half-wave (first-index tie-break)
#pragma unroll
  for (int off = 1; off <= 8; off <<= 1) {
#pragma unroll
    for (int v = 0; v < 8; ++v) {
      float ob = __shfl_xor(best[v], off, 32);
      int   oi = __shfl_xor(bidx[v], off, 32);
      if (ob < best[v] || (ob == best[v] && oi < bidx[v])) { best[v] = ob; bidx[v] = oi; }
    }
  }

  if (lo == 0) {
#pragma unroll
    for (int v = 0; v < 8; ++v) {
      int rl = rt * 16 + v + 8 * hi;                 // local row 0..63
      sR[(rl * 2 + ch) * 2 + 0] = best[v];
      sR[(rl * 2 + ch) * 2 + 1] = __int_as_float(bidx[v]);
    }
  }
  __syncthreads();
  if (tid < WG_ROWS) {                               // combine the two column halves
    float m0 = sR[(tid * 2 + 0) * 2];  int i0 = __float_as_int(sR[(tid * 2 + 0) * 2 + 1]);
    float m1 = sR[(tid * 2 + 1) * 2];  int i1 = __float_as_int(sR[(tid * 2 + 1) * 2 + 1]);
    outIdx[row0 + tid] = (m1 < m0 || (m1 == m0 && i1 < i0)) ? i1 : i0;
  }
}

// ---------------------------------------------------------------------------
// GEMM2: sim = emb[idx] @ emb^T with fused online logsumexp, self-mask and
// capture of sim[i, y_true[i]].  Same TDM-fed WMMA tiling.
// ---------------------------------------------------------------------------
__global__ __launch_bounds__(256, 1)
void vq_sim_lse(const float* __restrict__ emb,
                const int* __restrict__ idx,
                const int* __restrict__ ytrue,
                float* __restrict__ dterm)
{
  extern __shared__ float smem[];
  float* sA  = smem;
  float* sB0 = smem + WG_ROWS * LSTR;
  float* sB1 = sB0 + PANEL * LSTR;
  float* sR  = sB1 + PANEL * LSTR;

  const int tid  = threadIdx.x;
  const int lane = tid & 31;
  const int wave = tid >> 5;
  const int lo   = lane & 15;
  const int hi   = lane >> 4;
  const int rt   = wave >> 1;
  const int ch   = wave &  1;
  const int row0 = blockIdx.x * WG_ROWS;

  if (wave == 0) tdm_load_panel(sB0, emb, PANEL);

  if (tid < WG_ROWS) sR[tid] = __int_as_float(idx[row0 + tid]);
  __syncthreads();
  for (int t = tid; t < WG_ROWS * (E_DIM / 4); t += 256) {   // gather A = emb[idx]
    int r  = t >> 7;
    int c4 = t & 127;
    int code = __float_as_int(sR[r]);
    float4 v = *(const float4*)(emb + (size_t)code * E_DIM + c4 * 4);
    *(float4*)(sA + r * LSTR + c4 * 4) = v;
  }

  int   selfI[8], yI[8];
  float m[8], s[8], sy[8];
#pragma unroll
  for (int v = 0; v < 8; ++v) {
    int row = row0 + rt * 16 + v + 8 * hi;
    selfI[v] = idx[row];
    yI[v]    = ytrue[row];
    m[v] = -INFINITY; s[v] = 0.0f; sy[v] = -3.4e38f;
  }

  const float* aBase  = sA  + (rt * 16 + lo) * LSTR + 2 * hi;
  const float* bBase0 = sB0 + (ch * 16 + lo) * LSTR + 2 * hi;
  const float* bBase1 = sB1 + (ch * 16 + lo) * LSTR + 2 * hi;

  if (wave == 0) __builtin_amdgcn_s_wait_tensorcnt(0);
  __syncthreads();

  int cur = 0;
  for (int cb = 0; cb < N_E; cb += PANEL) {
    if (wave == 0 && cb + PANEL < N_E)
      tdm_load_panel(cur ? (void*)sB0 : (void*)sB1,
                     emb + (size_t)(cb + PANEL) * E_DIM, PANEL);

    const float* bB = cur ? bBase1 : bBase0;
    v8f acc0 = {}; v8f acc1 = {};
#pragma unroll 8
    for (int kk = 0; kk < E_DIM; kk += 8) {
      const int bo = kk + ((kk >= 256) ? 2 : 0);
      v2f a0 = *(const v2f*)(aBase + kk);
      v2f b0 = *(const v2f*)(bB + bo);
      v2f a1 = *(const v2f*)(aBase + kk + 4);
      v2f b1 = *(const v2f*)(bB + bo + 4);
      acc0 = __builtin_amdgcn_wmma_f32_16x16x4_f32(false, a0, false, b0,
                                                   (short)0, acc0, false, false);
      acc1 = __builtin_amdgcn_wmma_f32_16x16x4_f32(false, a1, false, b1,
                                                   (short)0, acc1, false, false);
    }

    const int col = cb + ch * 16 + lo;
#pragma unroll
    for (int v = 0; v < 8; ++v) {
      float val = acc0[v] + acc1[v];
      if (col == selfI[v]) val -= BIG_NEG;     // self-similarity mask
      if (col == yI[v])    sy[v] = val;        // capture sim at positive
      if (val > m[v]) { s[v] = s[v] * __expf(m[v] - val) + 1.0f; m[v] = val; }
      else            { s[v] += __expf(val - m[v]); }
    }

    if (wave == 0) __builtin_amdgcn_s_wait_tensorcnt(0);
    __syncthreads();
    cur ^= 1;
  }

  // merge softmax state across the 16 lanes of each half-wave
#pragma unroll
  for (int off = 1; off <= 8; off <<= 1) {
#pragma unroll
    for (int v = 0; v < 8; ++v) {
      float om = __shfl_xor(m[v], off, 32);
      float os = __shfl_xor(s[v], off, 32);
      float oy = __shfl_xor(sy[v], off, 32);
      if (om > m[v]) { s[v] = s[v] * __expf(m[v] - om) + os; m[v] = om; }
      else           { s[v] += os * __expf(om - m[v]); }
      sy[v] = fmaxf(sy[v], oy);
    }
  }

  __syncthreads();
  if (lo == 0) {
#pragma unroll
    for (int v = 0; v < 8; ++v) {
      int rl = rt * 16 + v + 8 * hi;
      sR[(rl * 2 + ch) * 3 + 0] = m[v];
      sR[(rl * 2 + ch) * 3 + 1] = s[v];
      sR[(rl * 2 + ch) * 3 + 2] = sy[v];
    }
  }
  __syncthreads();
  if (tid < WG_ROWS) {
    float m0 = sR[(tid * 2 + 0) * 3], s0 = sR[(tid * 2 + 0) * 3 + 1], y0 = sR[(tid * 2 + 0) * 3 + 2];
    float m1 = sR[(tid * 2 + 1) * 3], s1 = sR[(tid * 2 + 1) * 3 + 1], y1 = sR[(tid * 2 + 1) * 3 + 2];
    float M = fmaxf(m0, m1);
    float S = s0 * __expf(m0 - M) + s1 * __expf(m1 - M);
    float lse  = M + __logf(S);
    float simy = fmaxf(y0, y1);
    dterm[row0 + tid] = simy - lse;            // logp[i, y_i]
  }
}

// ---------------------------------------------------------------------------
// helpers
// ---------------------------------------------------------------------------
__global__ void row_norm(const float* __restrict__ mat, float* __restrict__ out, int nrows)
{
  int wave = threadIdx.x >> 5, lane = threadIdx.x & 31;
  int row = blockIdx.x * 8 + wave;
  if (row >= nrows) return;
  const float4* p = (const float4*)(mat + (size_t)row * E_DIM);
  float s = 0.0f;
#pragma unroll
  for (int q = 0; q < 4; ++q) {
    float4 v = p[lane + q * 32];
    s += v.x * v.x + v.y * v.y + v.z * v.z + v.w * v.w;
  }
#pragma unroll
  for (int off = 16; off > 0; off >>= 1) s += __shfl_xor(s, off, 32);
  if (lane == 0) out[row] = s;
}

__device__ __forceinline__ float gumbel_hash(unsigned i, unsigned j)
{
  unsigned h = i * 0x9E3779B9u + j * 0x85EBCA6Bu + 42u * 0xC2B2AE35u;
  h ^= h >> 16; h *= 0x7FEB352Du; h ^= h >> 15; h *= 0x846CA68Bu; h ^= h >> 16;
  float u = ((float)(h >> 8) + 0.5f) * (1.0f / 16777216.0f);
  return -__logf(-__logf(u));
}

__global__ void pick_positive(const int* __restrict__ label,
                              const int* __restrict__ idx,
                              int* __restrict__ ytrue)
{
  int wave = threadIdx.x >> 5, lane = threadIdx.x & 31;
  int row = blockIdx.x * 8 + wave;
  if (row >= N_ROWS) return;
  int self = idx[row];
  int cl   = label[self];
  float bg = -INFINITY; int bj = 0;
  for (int j = lane; j < N_E; j += 32) {
    if (j != self && label[j] == cl) {
      float g = gumbel_hash((unsigned)row, (unsigned)j);
      if (g > bg) { bg = g; bj = j; }
    }
  }
#pragma unroll
  for (int off = 16; off > 0; off >>= 1) {
    float og = __shfl_xor(bg, off, 32);
    int   oj = __shfl_xor(bj, off, 32);
    if (og > bg || (og == bg && oj < bj)) { bg = og; bj = oj; }
  }
  if (lane == 0) ytrue[row] = bj;
}

__global__ void gather_mse(const float* __restrict__ latent,
                           const float* __restrict__ emb,
                           const int* __restrict__ idx,
                           float* __restrict__ outQ,
                           float* __restrict__ mseAcc)
{
  __shared__ float red[8];
  float s = 0.0f;
  const size_t total4 = (size_t)N_ROWS * (E_DIM / 4);
  for (size_t t = (size_t)blockIdx.x * blockDim.x + threadIdx.x; t < total4;
       t += (size_t)gridDim.x * blockDim.x) {
    size_t r  = t >> 7;
    size_t c4 = t & 127;
    int code = idx[r];
    float4 e = *(const float4*)(emb + (size_t)code * E_DIM + c4 * 4);
    float4 z = *(const float4*)(latent + r * E_DIM + c4 * 4);
    *(float4*)(outQ + r * E_DIM + c4 * 4) = e;   // forward STE output == x_q
    float dx = e.x - z.x, dy = e.y - z.y, dz = e.z - z.z, dw = e.w - z.w;
    s += dx * dx + dy * dy + dz * dz + dw * dw;
  }
#pragma unroll
  for (int off = 16; off > 0; off >>= 1) s += __shfl_xor(s, off, 32);
  if ((threadIdx.x & 31) == 0) red[threadIdx.x >> 5] = s;
  __syncthreads();
  if (threadIdx.x == 0) {
    float t = 0.0f;
    for (int w = 0; w < 8; ++w) t += red[w];
    atomicAdd(mseAcc, t);
  }
}

__global__ void write_indices(const int* __restrict__ idx, float* __restrict__ outI)
{
  int g = blockIdx.x * blockDim.x + threadIdx.x;
  if (g < N_ROWS) outI[g] = (float)idx[g];
}

__global__ void zero_scalar(float* p)
{
  if (threadIdx.x == 0 && blockIdx.x == 0) *p = 0.0f;
}

__global__ void finalize_loss(const float* __restrict__ dterm,
                              const float* __restrict__ mseAcc,
                              float* __restrict__ outLoss)
{
  __shared__ float red[8];
  float s = 0.0f;
  for (int t = threadIdx.x; t < N_ROWS; t += 256) s += dterm[t];
#pragma unroll
  for (int off = 16; off > 0; off >>= 1) s += __shfl_xor(s, off, 32);
  if ((threadIdx.x & 31) == 0) red[threadIdx.x >> 5] = s;
  __syncthreads();
  if (threadIdx.x == 0) {
    float d = 0.0f;
    for (int w = 0; w < 8; ++w) d += red[w];
    float diversity = -(d / (float)N_ROWS);
    float mse = *mseAcc / (float)((size_t)N_ROWS * E_DIM);
    // codebook_loss + MU*commitment_loss (equal in forward) + BETA*diversity
    *outLoss = (1.0f + MU) * mse + BETA * diversity;
  }
}

// ---------------------------------------------------------------------------
extern "C" void kernel_launch(void* const* d_in, const int* in_sizes, int n_in,
                              void* d_out, int out_size, void* d_ws, size_t ws_size,
                              hipStream_t stream)
{
  const float* latent = (const float*)d_in[0];   // x: [64,128,512] -> [8192,512]
  const float* emb    = (const float*)d_in[1];   // [8192,512]
  const int*   label  = (const int*)d_in[2];     // [8192]
  // d_in[3] = idx scalar: deterministic branch, unused

  char* ws = (char*)d_ws;
  int*   dIdx  = (int*)ws;    ws += N_ROWS * sizeof(int);
  int*   dY    = (int*)ws;    ws += N_ROWS * sizeof(int);
  float* dEn   = (float*)ws;  ws += N_E * sizeof(float);
  float* dTerm = (float*)ws;  ws += N_ROWS * sizeof(float);
  float* dMse  = (float*)ws;  ws += sizeof(float);

  float* outQ    = (float*)d_out;                      // 4194304 floats
  float* outLoss = outQ + (size_t)N_ROWS * E_DIM;      // 1 float
  float* outIdx  = outLoss + 1;                        // 8192 floats

  hipFuncSetAttribute(reinterpret_cast<const void*>(vq_dist_argmin),
                      hipFuncAttributeMaxDynamicSharedMemorySize, SMEM_BYTES);
  hipFuncSetAttribute(reinterpret_cast<const void*>(vq_sim_lse),
                      hipFuncAttributeMaxDynamicSharedMemorySize, SMEM_BYTES);

  zero_scalar<<<1, 64, 0, stream>>>(dMse);
  row_norm<<<N_E / 8, 256, 0, stream>>>(emb, dEn, N_E);
  vq_dist_argmin<<<N_ROWS / WG_ROWS, 256, SMEM_BYTES, stream>>>(latent, emb, dEn, dIdx);
  pick_positive<<<N_ROWS / 8, 256, 0, stream>>>(label, dIdx, dY);
  vq_sim_lse<<<N_ROWS / WG_ROWS, 256, SMEM_BYTES, stream>>>(emb, dIdx, dY, dTerm);
  gather_mse<<<2048, 256, 0, stream>>>(latent, emb, dIdx, outQ, dMse);
  write_indices<<<(N_ROWS + 255) / 256, 256, 0, stream>>>(dIdx, outIdx);
  finalize_loss<<<1, 256, 0, stream>>>(dTerm, dMse, outLoss);
}